// BinaryRegionsProposal_46471546142906
// MI455X (gfx1250) — compile-verified
//
#include <hip/hip_runtime.h>

#define NIMG   512
#define NPIX   (224*224)   // 50176
#define NBINS  256
#define NWAVES 8
#define CHF    4096        // floats per staging buffer (16 KB); 12 full chunks
#define NFULL  12          // 12 * 4096 = 49152
#define TAIL   (NPIX - NFULL*CHF)   // 1024 tail floats, done with direct loads

typedef float v2f __attribute__((ext_vector_type(2)));
typedef float v4f __attribute__((ext_vector_type(4)));
typedef float v8f __attribute__((ext_vector_type(8)));

// ---------------------------------------------------------------------------
// Kernel 1: per-image 256-bin histogram.
// Double-buffered async global->LDS DMA staging (ASYNCcnt pipelined: issue
// chunk c+1, wait <=4 for chunk c since async loads complete in order), then
// bin out of LDS into per-wave privatized LDS histograms (ds_add_u32).
// One 256-thread block (8 waves) per image.
// ---------------------------------------------------------------------------
__global__ void __launch_bounds__(256) hist_kernel(const float* __restrict__ x,
                                                   unsigned* __restrict__ hist) {
  // Single LDS allocation -> assumed to start at LDS byte offset 0
  // (the async-to-LDS instruction takes a raw LDS byte offset).
  __shared__ __align__(16) unsigned shmem[NWAVES * NBINS + 2 * CHF];
  const int tid  = threadIdx.x;
  const int wave = tid >> 5;     // wave32
  const int img  = blockIdx.x;

  unsigned* lh    = shmem;                                        // 8 sub-hists
  float*    stage = reinterpret_cast<float*>(shmem + NWAVES * NBINS);
  const unsigned sbase = NWAVES * NBINS * 4u;                     // byte offset

  for (int i = tid; i < NWAVES * NBINS; i += 256) lh[i] = 0u;
  __syncthreads();   // hist zeroed before any ds_add

  const float* base = x + (size_t)img * NPIX;
  unsigned* h = lh + wave * NBINS;

  // prologue: issue chunk 0 into buffer 0 (4 x b128 per thread)
#pragma unroll
  for (int r = 0; r < 4; ++r) {
    const int i = tid + r * 256;                     // float4 index in chunk
    unsigned     ldsoff = sbase + ((unsigned)i << 4);
    const float* gp     = base + (i << 2);
    asm volatile("global_load_async_to_lds_b128 %0, %1, off"
                 :: "v"(ldsoff), "v"(gp) : "memory");
  }

  int cur = 0;
  for (int c = 0; c < NFULL; ++c) {
    // issue next chunk into the other buffer (its reads finished last iter)
    if (c + 1 < NFULL) {
      const unsigned nb = sbase + (unsigned)(cur ^ 1) * (CHF * 4u);
      const float*   np = base + (c + 1) * CHF;
#pragma unroll
      for (int r = 0; r < 4; ++r) {
        const int i = tid + r * 256;
        unsigned     ldsoff = nb + ((unsigned)i << 4);
        const float* gp     = np + (i << 2);
        asm volatile("global_load_async_to_lds_b128 %0, %1, off"
                     :: "v"(ldsoff), "v"(gp) : "memory");
      }
      asm volatile("s_wait_asynccnt 0x4" ::: "memory");  // chunk c resident
    } else {
      asm volatile("s_wait_asynccnt 0x0" ::: "memory");
    }
    __syncthreads();

    const float* st = stage + cur * CHF;
#pragma unroll
    for (int r = 0; r < CHF / 256; ++r) {            // 16 floats per thread
      int b = (int)floorf(st[tid + r * 256] * 255.0f);
      b = b < 0 ? 0 : (b > 255 ? 255 : b);
      atomicAdd(&h[b], 1u);                          // ds_add_u32
    }
    __syncthreads();   // chunk fully consumed before its buffer is refilled
    cur ^= 1;
  }

  // tail (1024 floats): direct global loads, no staging
#pragma unroll
  for (int r = 0; r < TAIL / 256; ++r) {
    int b = (int)floorf(base[NFULL * CHF + tid + r * 256] * 255.0f);
    b = b < 0 ? 0 : (b > 255 ? 255 : b);
    atomicAdd(&h[b], 1u);
  }
  __syncthreads();

  // tid covers exactly the 256 bins
  unsigned s = 0;
  for (int w = 0; w < NWAVES; ++w) s += lh[w * NBINS + tid];
  hist[img * NBINS + tid] = s;
}

// ---------------------------------------------------------------------------
// 256-element inclusive scan via WMMA: view v as 16x16 matrix H.
//   pass 1:  R = H x U          (U upper-triangular ones)  -> row-wise scans
//   pass 2:  D = Ls x Trep + R  (Ls strictly-lower ones, Trep[r][*]=rowtotal r)
// Each pass = 4 chained V_WMMA_F32_16X16X4_F32 (K accumulation).
// Wave32, EXEC all ones. Layouts per ISA 7.12.2.
// ---------------------------------------------------------------------------
__device__ __forceinline__ v8f scan256_wmma(const float* sv, float* sT, int lane) {
  const int m    = lane & 15;            // A row / B,C,D column
  const int koff = (lane >> 4) << 1;     // lane half selects K {0,1} vs {2,3}
  v8f c = {};
#pragma unroll
  for (int k = 0; k < 4; ++k) {
    const int j0 = 4 * k + koff;
    v2f a, b;
    a.x = sv[m * 16 + j0];               // A = H[:, 4k..4k+3]
    a.y = sv[m * 16 + j0 + 1];
    b.x = (j0     <= m) ? 1.0f : 0.0f;   // B = U[4k..4k+3, :]
    b.y = (j0 + 1 <= m) ? 1.0f : 0.0f;
    c = __builtin_amdgcn_wmma_f32_16x16x4_f32(false, a, false, b, (short)0, c,
                                              false, false);
  }
  // Row totals T[r] = R[r][15] live in lanes 15 (rows 0..7) and 31 (rows 8..15)
  if (m == 15) {
    const int rbase = (lane >> 4) * 8;
#pragma unroll
    for (int j = 0; j < 8; ++j) sT[rbase + j] = c[j];
  }
  __syncthreads();
#pragma unroll
  for (int k = 0; k < 4; ++k) {
    const int j0 = 4 * k + koff;
    v2f a, b;
    a.x = (j0     < m) ? 1.0f : 0.0f;    // A = Ls[:, 4k..4k+3]
    a.y = (j0 + 1 < m) ? 1.0f : 0.0f;
    b.x = sT[j0];                        // B = Trep[4k..4k+3, :]
    b.y = sT[j0 + 1];
    c = __builtin_amdgcn_wmma_f32_16x16x4_f32(false, a, false, b, (short)0, c,
                                              false, false);
  }
  return c;
}

// ---------------------------------------------------------------------------
// Kernel 2: Otsu threshold per image. One wave (32 threads) per image.
// ---------------------------------------------------------------------------
__global__ void __launch_bounds__(32) otsu_kernel(const unsigned* __restrict__ hist,
                                                  float* __restrict__ thresh) {
  __shared__ float sv[NBINS], sc[NBINS], sw[NBINS], ss[NBINS], sT[16];
  const int lane = threadIdx.x;
  const int img  = blockIdx.x;

  const unsigned* h = hist + img * NBINS;
  for (int i = lane; i < NBINS; i += 32) {
    float f = (float)h[i];
    sv[i] = f;                  // hist
    sc[i] = f * (float)i;       // hist * bin_center
  }
  __syncthreads();

  v8f w1 = scan256_wmma(sv, sT, lane);   // cumsum(hist)
  __syncthreads();
  v8f s1 = scan256_wmma(sc, sT, lane);   // cumsum(hist * center)

  // Scatter C/D layout back to linear bins: element (row = j+8*half, col = lane&15)
  const int n     = lane & 15;
  const int rbase = (lane >> 4) * 8;
#pragma unroll
  for (int j = 0; j < 8; ++j) {
    sw[(rbase + j) * 16 + n] = w1[j];
    ss[(rbase + j) * 16 + n] = s1[j];
  }
  __syncthreads();

  const float total = sw[NBINS - 1];     // == NPIX
  const float stot  = ss[NBINS - 1];

  // vmin / vmax = first / last non-empty bin
  int lmin = 256, lmax = -1;
  for (int i = lane; i < NBINS; i += 32)
    if (sv[i] > 0.0f) { lmin = min(lmin, i); lmax = max(lmax, i); }
#pragma unroll
  for (int off = 16; off > 0; off >>= 1) {
    lmin = min(lmin, __shfl_xor(lmin, off, 32));
    lmax = max(lmax, __shfl_xor(lmax, off, 32));
  }

  // argmax of inter-class variance over valid t in [vmin, vmax), first-max wins
  float bestv = -2.0f; int besti = 0x7fffffff;
  for (int i = lane; i < NBINS; i += 32) {
    float var = -1.0f;
    if (i >= lmin && i < lmax) {
      float w1v = sw[i];
      float w2v = total - w1v;
      float s1v = ss[i];
      float m1  = s1v / w1v;
      float m2  = (stot - s1v) / w2v;
      float d   = m1 - m2;
      var = w1v * w2v * d * d;
    }
    if (var > bestv || (var == bestv && i < besti)) { bestv = var; besti = i; }
  }
#pragma unroll
  for (int off = 16; off > 0; off >>= 1) {
    float ov = __shfl_xor(bestv, off, 32);
    int   oi = __shfl_xor(besti, off, 32);
    if (ov > bestv || (ov == bestv && oi < besti)) { bestv = ov; besti = oi; }
  }

  if (lane == 0) {
    float th = (float)besti;
    if (besti == 0)   th = 1.0f;
    if (besti == 255) th = 254.0f;
    if (lmin == lmax) th = 1.0e9f;   // bad image: cam > th never true -> all zeros
    thresh[img] = th;
  }
}

// ---------------------------------------------------------------------------
// Kernel 3: binarize.  out = (floor(x*255) > th) ? 1 : 0   (float output)
// Reads hit L2 (input was just streamed by the histogram pass and fits in the
// 192 MB L2); stores are non-temporal so the 102.8 MB output doesn't evict it.
// Uses clang ext-vector v4f (b128 load / b128 NT store).
// ---------------------------------------------------------------------------
__global__ void __launch_bounds__(256) binarize_kernel(const float* __restrict__ x,
                                                       const float* __restrict__ thresh,
                                                       float* __restrict__ out) {
  const int img = blockIdx.y;
  const int i4  = blockIdx.x * 256 + threadIdx.x;
  const float th = thresh[img];
  const v4f* p = reinterpret_cast<const v4f*>(x + (size_t)img * NPIX);
  v4f v = p[i4];
  v4f r;
  r.x = (floorf(v.x * 255.0f) > th) ? 1.0f : 0.0f;
  r.y = (floorf(v.y * 255.0f) > th) ? 1.0f : 0.0f;
  r.z = (floorf(v.z * 255.0f) > th) ? 1.0f : 0.0f;
  r.w = (floorf(v.w * 255.0f) > th) ? 1.0f : 0.0f;
  v4f* q = reinterpret_cast<v4f*>(out + (size_t)img * NPIX) + i4;
  __builtin_nontemporal_store(r, q);     // NT store policy
}

// ---------------------------------------------------------------------------
extern "C" void kernel_launch(void* const* d_in, const int* in_sizes, int n_in,
                              void* d_out, int out_size, void* d_ws, size_t ws_size,
                              hipStream_t stream) {
  const float* x   = (const float*)d_in[0];
  float*       out = (float*)d_out;

  unsigned* hist   = (unsigned*)d_ws;                                  // 512 KB
  float*    thresh = (float*)((char*)d_ws + (size_t)NIMG * NBINS * 4); //   2 KB

  hipLaunchKernelGGL(hist_kernel,     dim3(NIMG),              dim3(256), 0, stream, x, hist);
  hipLaunchKernelGGL(otsu_kernel,     dim3(NIMG),              dim3(32),  0, stream, hist, thresh);
  hipLaunchKernelGGL(binarize_kernel, dim3(NPIX / 1024, NIMG), dim3(256), 0, stream, x, thresh, out);
}